// NMRModelTorch_89567247990814
// MI455X (gfx1250) — compile-verified
//
#include <hip/hip_runtime.h>
#include <math.h>

#define DIMN   128
#define MSZ    (DIMN*DIMN)
#define NSP    7
#define NSAMP  4096
#define CHUNKS 64
#define CHLEN  64
#define DT_F     1e-5f
#define B0_F     80.0f
#define T2_F     1.0f
#define SW_F     1000.0f
#define TWO_PI_F 6.28318530717958647692f
#define PI_F     3.14159265358979323846f

typedef float v2f __attribute__((ext_vector_type(2)));
typedef float v8f __attribute__((ext_vector_type(8)));
typedef unsigned int u32x4 __attribute__((ext_vector_type(4)));
typedef unsigned int u32x8 __attribute__((ext_vector_type(8)));

static __device__ __forceinline__ v8f v8zero() {
  v8f z = {0.f,0.f,0.f,0.f,0.f,0.f,0.f,0.f};
  return z;
}

// One K=4 slab of a 16x16 f32 WMMA tile: acc += (NEGA ? -A : A) * (TRANSB ? B^T : B)
// A,B are 128x128 row-major (global or LDS). Lane layout per CDNA5 ISA 7.12.2.
template<bool NEGA, bool TRANSB>
static __device__ __forceinline__ v8f wmma_k4(const float* __restrict__ A,
                                              const float* __restrict__ B,
                                              int mBase, int nBase, int k0, v8f acc)
{
  const int lane = threadIdx.x & 31;
  const int half = lane >> 4;       // 0: K pair {k0,k0+1}, 1: {k0+2,k0+3}
  const int l16  = lane & 15;
  const int kA   = k0 + (half << 1);
  v2f a, b;
  a[0] = A[(mBase + l16) * DIMN + kA];
  a[1] = A[(mBase + l16) * DIMN + kA + 1];
  if (NEGA) { a[0] = -a[0]; a[1] = -a[1]; }   // f32 WMMA NEG bits must be 0 per ISA -> negate in VALU
  if (TRANSB) {
    b[0] = B[(nBase + l16) * DIMN + kA];
    b[1] = B[(nBase + l16) * DIMN + kA + 1];
  } else {
    b[0] = B[ kA      * DIMN + nBase + l16];
    b[1] = B[(kA + 1) * DIMN + nBase + l16];
  }
  return __builtin_amdgcn_wmma_f32_16x16x4_f32(false, a, false, b, (short)0, acc, false, false);
}

static __device__ __forceinline__ void store_tile(float* D, int mBase, int nBase, v8f acc)
{
  const int lane = threadIdx.x & 31;
  const int half = lane >> 4, l16 = lane & 15;
#pragma unroll
  for (int v = 0; v < 8; ++v)
    D[(mBase + v + 8*half) * DIMN + nBase + l16] = acc[v];
}

static __device__ __forceinline__ void store_tile_ab(float* D, int mBase, int nBase,
                                                     v8f acc, float alpha, float beta)
{
  const int lane = threadIdx.x & 31;
  const int half = lane >> 4, l16 = lane & 15;
#pragma unroll
  for (int v = 0; v < 8; ++v) {
    const int idx = (mBase + v + 8*half) * DIMN + nBase + l16;
    float d = alpha * acc[v];
    if (beta != 0.f) d += beta * D[idx];
    D[idx] = d;
  }
}

// Issue a 2D (128x128 f32) TENSOR_LOAD_TO_LDS via the Tensor Data Mover.
// D# group0: count=1 | lds_addr | 57-bit global addr | type=2 ("image")
// D# group1: data_size=4B, tensor 128x128, tile 128x128, dim0_stride=128, no multicast.
static __device__ __forceinline__ void tdm_load_128x128(unsigned lds_byte_off,
                                                        unsigned long long gaddr)
{
  u32x4 g0;
  u32x8 g1;
  g0[0] = 1u;                                            // count = 1 valid descriptor
  g0[1] = lds_byte_off;                                  // lds_addr (bytes)
  g0[2] = (unsigned)(gaddr & 0xFFFFFFFFull);             // global_addr[31:0]
  g0[3] = (unsigned)((gaddr >> 32) & 0x01FFFFFFull)      // global_addr[56:32]
        | (2u << 30);                                    // type = 2
  g1[0] = (2u << 16);                                    // workgroup_mask=0, data_size=2 (4 bytes)
  g1[1] = ((unsigned)DIMN & 0xFFFFu) << 16;              // tensor_dim0[15:0] @ bits 63:48
  g1[2] = ((unsigned)DIMN & 0xFFFFu) << 16;              // dim0[31:16]=0 | tensor_dim1[15:0]
  g1[3] = ((unsigned)DIMN) << 16;                        // dim1[31:16]=0 | tile_dim0 = 128
  g1[4] = (unsigned)DIMN;                                // tile_dim1 = 128, tile_dim2 = 0
  g1[5] = (unsigned)DIMN;                                // tensor_dim0_stride[31:0] = 128 elems
  g1[6] = 0u;                                            // stride hi | tensor_dim1_stride lo
  g1[7] = 0u;
  asm volatile("tensor_load_to_lds %0, %1" :: "s"(g0), "s"(g1) : "memory");
}

// ---------------- build X = DT*H0 (bit-trick Hamiltonian) and rho0 = IHx ----------------
__global__ void build_kernel(float* __restrict__ X, float* __restrict__ rho0,
                             const float* __restrict__ h)
{
  const int t = blockIdx.x * blockDim.x + threadIdx.x;   // 16384
  const int a = t >> 7, b = t & 127;
  float h0 = 0.f;
  if (a == b) {
    for (int i = 0; i < NSP; ++i) {
      const int p = NSP - 1 - i;                          // spin i <-> bit (MSB-first Kron order)
      const float szi = ((a >> p) & 1) ? -0.5f : 0.5f;
      h0 += TWO_PI_F * B0_F * h[i*NSP + i] * szi;
      for (int j = i + 1; j < NSP; ++j) {
        const int q = NSP - 1 - j;
        const float szj = ((a >> q) & 1) ? -0.5f : 0.5f;
        h0 += TWO_PI_F * h[i*NSP + j] * szi * szj;        // IzIz coupling (diagonal)
      }
    }
  } else {
    const int x = a ^ b;
    if (__popc(x) == 2) {                                 // flip-flop (IxIx+IyIy) term
      const int pLo = __ffs(x) - 1;
      const int pHi = 31 - __clz(x);
      const int i = NSP - 1 - pHi, j = NSP - 1 - pLo;     // i < j  (upper triangle of h)
      if (((a >> pHi) & 1) != ((a >> pLo) & 1))
        h0 += TWO_PI_F * h[i*NSP + j] * 0.5f;
    }
  }
  X[t]    = DT_F * h0;
  rho0[t] = (__popc(a ^ b) == 1) ? 0.5f : 0.f;            // rho0 = U90y IHz U90y^dag = IHx
}

// ---------------- generic 128x128 f32 matmul: D = alpha*A*B + beta*D ----------------
__global__ __launch_bounds__(256) void mm128_kernel(float* __restrict__ D,
                                                    const float* __restrict__ A,
                                                    const float* __restrict__ B,
                                                    float alpha, float beta)
{
  const int wave = blockIdx.x * (blockDim.x >> 5) + (threadIdx.x >> 5);  // 8 blocks x 8 waves = 64 tiles
  const int mB = (wave >> 3) << 4, nB = (wave & 7) << 4;
  v8f acc = v8zero();
  for (int k0 = 0; k0 < DIMN; k0 += 4)
    acc = wmma_k4<false,false>(A, B, mB, nB, k0, acc);
  store_tile_ab(D, mB, nB, acc, alpha, beta);
}

// ---------------- complex 128x128 matmul: (DR+iDI) = (AR+iAI)(BR+iBI) ----------------
__global__ __launch_bounds__(256) void cmm128_kernel(float* __restrict__ DR, float* __restrict__ DI,
                                                     const float* __restrict__ AR, const float* __restrict__ AI,
                                                     const float* __restrict__ BR, const float* __restrict__ BI)
{
  const int wave = blockIdx.x * (blockDim.x >> 5) + (threadIdx.x >> 5);
  const int mB = (wave >> 3) << 4, nB = (wave & 7) << 4;
  v8f r = v8zero(), im = v8zero();
  for (int k0 = 0; k0 < DIMN; k0 += 4) {
    r  = wmma_k4<false,false>(AR, BR, mB, nB, k0, r);
    r  = wmma_k4<true ,false>(AI, BI, mB, nB, k0, r);
    im = wmma_k4<false,false>(AR, BI, mB, nB, k0, im);
    im = wmma_k4<false,false>(AI, BR, mB, nB, k0, im);
  }
  store_tile(DR, mB, nB, r);
  store_tile(DI, mB, nB, im);
}

// ---------------- elementwise Taylor combine: C = cos-series, S0 = sinc-series ----------------
__global__ void build_cs_kernel(float* __restrict__ Cc, float* __restrict__ S0,
                                const float* __restrict__ X2, const float* __restrict__ X4,
                                const float* __restrict__ X6)
{
  const int t = blockIdx.x * blockDim.x + threadIdx.x;
  const float ide = ((t >> 7) == (t & 127)) ? 1.f : 0.f;
  const float x2 = X2[t], x4 = X4[t], x6 = X6[t];
  Cc[t] = ide - 0.5f*x2 + (1.f/24.f)*x4 - (1.f/720.f)*x6;        // cos(X)
  S0[t] = ide - (1.f/6.f)*x2 + (1.f/120.f)*x4 - (1.f/5040.f)*x6; // sin(X) = X*S0
}

__global__ void ident_kernel(float* __restrict__ QR0, float* __restrict__ QI0)
{
  const int t = blockIdx.x * blockDim.x + threadIdx.x;
  QR0[t] = ((t >> 7) == (t & 127)) ? 1.f : 0.f;
  QI0[t] = 0.f;
}

// ---------------- batched chain seeding: T = Q_c * rho0 ; rho_c = T * Q_c^dag ----------------
__global__ __launch_bounds__(256) void init1_kernel(float* __restrict__ TRb, float* __restrict__ TIb,
                                                    const float* __restrict__ QRb, const float* __restrict__ QIb,
                                                    const float* __restrict__ rho0)
{
  const int chain = blockIdx.y;
  const float* QR = QRb + (size_t)chain * MSZ;
  const float* QI = QIb + (size_t)chain * MSZ;
  float* TR = TRb + (size_t)chain * MSZ;
  float* TI = TIb + (size_t)chain * MSZ;
  const int wave = blockIdx.x * (blockDim.x >> 5) + (threadIdx.x >> 5);
  const int mB = (wave >> 3) << 4, nB = (wave & 7) << 4;
  v8f tr = v8zero(), ti = v8zero();
  for (int k0 = 0; k0 < DIMN; k0 += 4) {
    tr = wmma_k4<false,false>(QR, rho0, mB, nB, k0, tr);
    ti = wmma_k4<false,false>(QI, rho0, mB, nB, k0, ti);
  }
  store_tile(TR, mB, nB, tr);
  store_tile(TI, mB, nB, ti);
}

__global__ __launch_bounds__(256) void init2_kernel(float* __restrict__ Rb, float* __restrict__ Jb,
                                                    const float* __restrict__ TRb, const float* __restrict__ TIb,
                                                    const float* __restrict__ QRb, const float* __restrict__ QIb)
{
  const int chain = blockIdx.y;
  const float* TR = TRb + (size_t)chain * MSZ;
  const float* TI = TIb + (size_t)chain * MSZ;
  const float* QR = QRb + (size_t)chain * MSZ;
  const float* QI = QIb + (size_t)chain * MSZ;
  float* R = Rb + (size_t)chain * MSZ;
  float* J = Jb + (size_t)chain * MSZ;
  const int wave = blockIdx.x * (blockDim.x >> 5) + (threadIdx.x >> 5);
  const int mB = (wave >> 3) << 4, nB = (wave & 7) << 4;
  v8f r = v8zero(), j = v8zero();
  for (int k0 = 0; k0 < DIMN; k0 += 4) {
    // R = TR*QR^T + TI*QI^T ; J = TI*QR^T - TR*QI^T   (Q^dag = QR^T - i QI^T)
    r = wmma_k4<false,true>(TR, QR, mB, nB, k0, r);
    r = wmma_k4<false,true>(TI, QI, mB, nB, k0, r);
    j = wmma_k4<false,true>(TI, QR, mB, nB, k0, j);
    j = wmma_k4<true ,true>(TR, QI, mB, nB, k0, j);
  }
  store_tile(R, mB, nB, r);
  store_tile(J, mB, nB, j);
}

// ---------------- 64 parallel chains x 64 sequential steps of rho <- P rho P^dag ----------------
// P = PR + i*PI (both symmetric). rho = R + i*J kept in LDS (loaded by the Tensor Data Mover).
// Per step (8 real WMMA matmuls):
//   A1 = PR*R - PI*J ; A2 = PR*J + PI*R           (to global scratch)
//   R' = A1*PR + A2*PI ; J' = A2*PR - A1*PI       (back to LDS)
// trace(Op rho) recorded BEFORE the update (matches lax.scan semantics).
__global__ __launch_bounds__(256) void evolve_kernel(
    const float* __restrict__ PR, const float* __restrict__ PI,
    const float* __restrict__ Rg, const float* __restrict__ Jg,
    float* __restrict__ A1g, float* __restrict__ A2g,
    float* __restrict__ traces)
{
  extern __shared__ float lds[];               // R (16384) | J (16384) | red (512)
  float* Rs  = lds;
  float* Js  = lds + MSZ;
  float* red = lds + 2*MSZ;
  const int chain = blockIdx.x;
  const float* Rin = Rg + (size_t)chain * MSZ;
  const float* Jin = Jg + (size_t)chain * MSZ;
  float* A1 = A1g + (size_t)chain * MSZ;
  float* A2 = A2g + (size_t)chain * MSZ;

  __builtin_prefetch(PR, 0, 1);                // global_prefetch: P matrices are L2-resident & shared
  __builtin_prefetch(PI, 0, 1);

  // DMA rho into LDS via TDM (wave 0 issues; TENSORcnt tracks completion), others park at barrier.
  if (threadIdx.x < 32) {
    tdm_load_128x128(0u,                      (unsigned long long)(uintptr_t)Rin);
    tdm_load_128x128((unsigned)(MSZ * 4u),    (unsigned long long)(uintptr_t)Jin);
    __builtin_amdgcn_s_wait_tensorcnt(0);
  }
  __syncthreads();

  const int wave = threadIdx.x >> 5;

#pragma unroll 1
  for (int s = 0; s < CHLEN; ++s) {
    // ---- trace(Op rho): Op = IHx + i*IHy is sparse (one-bit-flip elements only) ----
    float tre = 0.f, tim = 0.f;
    const int a = threadIdx.x;
    if (a < DIMN) {
      for (int p = 0; p < NSP; ++p) {
        const int b = a ^ (1 << p);
        const float orv = 0.5f;                               // IHx element
        const float oiv = ((a >> p) & 1) ? 0.5f : -0.5f;      // Im(IHy) element
        const float rba = Rs[b*DIMN + a], jba = Js[b*DIMN + a];
        tre += orv*rba - oiv*jba;
        tim += orv*jba + oiv*rba;
      }
    }
    red[threadIdx.x]       = tre;
    red[256 + threadIdx.x] = tim;
    __syncthreads();
    if (threadIdx.x == 0) {
      float sr = 0.f, si = 0.f;
      for (int t = 0; t < DIMN; ++t) { sr += red[t]; si += red[256 + t]; }
      const int k = chain * CHLEN + s;
      traces[2*k]     = sr;
      traces[2*k + 1] = si;
    }
    __syncthreads();

    // ---- phase 1: A1/A2 to global scratch ----
#pragma unroll 1
    for (int t8 = 0; t8 < 8; ++t8) {
      const int tile = wave * 8 + t8;
      const int mB = (tile >> 3) << 4, nB = (tile & 7) << 4;
      v8f a1 = v8zero(), a2 = v8zero();
      for (int k0 = 0; k0 < DIMN; k0 += 4) {
        a1 = wmma_k4<false,false>(PR, Rs, mB, nB, k0, a1);
        a1 = wmma_k4<true ,false>(PI, Js, mB, nB, k0, a1);
        a2 = wmma_k4<false,false>(PR, Js, mB, nB, k0, a2);
        a2 = wmma_k4<false,false>(PI, Rs, mB, nB, k0, a2);
      }
      store_tile(A1, mB, nB, a1);
      store_tile(A2, mB, nB, a2);
    }
    __threadfence();
    __syncthreads();

    // ---- phase 2: new R/J back into LDS ----
#pragma unroll 1
    for (int t8 = 0; t8 < 8; ++t8) {
      const int tile = wave * 8 + t8;
      const int mB = (tile >> 3) << 4, nB = (tile & 7) << 4;
      v8f r = v8zero(), j = v8zero();
      for (int k0 = 0; k0 < DIMN; k0 += 4) {
        r = wmma_k4<false,false>(A1, PR, mB, nB, k0, r);
        r = wmma_k4<false,false>(A2, PI, mB, nB, k0, r);
        j = wmma_k4<false,false>(A2, PR, mB, nB, k0, j);
        j = wmma_k4<true ,false>(A1, PI, mB, nB, k0, j);
      }
      store_tile(Rs, mB, nB, r);
      store_tile(Js, mB, nB, j);
    }
    __syncthreads();
  }
}

// ---------------- FID + linspaces ----------------
__global__ void post_kernel(const float* __restrict__ traces, float* __restrict__ dout)
{
  const int t = blockIdx.x * blockDim.x + threadIdx.x;     // 8192 threads
  if (t < NSAMP) {
    const float ap = expf((-DT_F / T2_F) * (float)t);
    dout[2*t]     = traces[2*t]     * ap;                  // FID (interleaved complex)
    dout[2*t + 1] = traces[2*t + 1] * ap;
    const float AQ = (float)NSAMP / SW_F;
    dout[2*NSAMP + t] = AQ * (float)t / (float)(NSAMP - 1); // time_series
  }
  // freq_series = linspace(-SW/2, SW/2, 8192)
  dout[2*NSAMP + NSAMP + 2*(2*NSAMP) + t] =
      -0.5f*SW_F + SW_F * (float)t / (float)(2*NSAMP - 1);
}

// ---------------- single-block 8192-pt Stockham FFT in LDS, with fftshift ----------------
__global__ __launch_bounds__(256) void fft_kernel(const float* __restrict__ fid,
                                                  float* __restrict__ out)
{
  extern __shared__ float sm[];                // 2 x 8192 x float2 = 128 KB
  float2* A = (float2*)sm;
  float2* B = (float2*)(sm + 2*8192);
  const int NF = 2 * NSAMP;                    // 8192

  for (int t = threadIdx.x; t < NF; t += 256) {
    if (t < NSAMP) { A[t].x = fid[2*t]; A[t].y = fid[2*t + 1]; }
    else           { A[t].x = 0.f;      A[t].y = 0.f; }
  }
  __syncthreads();

  float2 *src = A, *dst = B;
  for (int Ns = 1; Ns < NF; Ns <<= 1) {
    for (int j = threadIdx.x; j < NF/2; j += 256) {
      const int k = j & (Ns - 1);
      const float2 u = src[j];
      const float2 v = src[j + NF/2];
      const float ang = -PI_F * (float)k / (float)Ns;   // = -2*pi*k/(2*Ns)
      const float cw = cosf(ang), sw = sinf(ang);
      float2 tv; tv.x = v.x*cw - v.y*sw; tv.y = v.x*sw + v.y*cw;
      const int idxD = ((j - k) << 1) + k;              // (j/Ns)*2*Ns + k
      dst[idxD].x      = u.x + tv.x; dst[idxD].y      = u.y + tv.y;
      dst[idxD + Ns].x = u.x - tv.x; dst[idxD + Ns].y = u.y - tv.y;
    }
    __syncthreads();
    float2* tmp = src; src = dst; dst = tmp;
  }
  // fftshift + (phase = exp(i*0) = 1)
  for (int t = threadIdx.x; t < NF; t += 256) {
    const float2 v = src[(t + NF/2) & (NF - 1)];
    out[2*t] = v.x; out[2*t + 1] = v.y;
  }
}

extern "C" void kernel_launch(void* const* d_in, const int* in_sizes, int n_in,
                              void* d_out, int out_size, void* d_ws, size_t ws_size,
                              hipStream_t stream)
{
  (void)in_sizes; (void)n_in; (void)out_size; (void)ws_size;
  const float* h_mat = (const float*)d_in[0];   // 7x7 f32
  float* w   = (float*)d_ws;
  float* out = (float*)d_out;

  // workspace layout (float offsets)
  float* X    = w + (size_t)0  * MSZ;
  float* X2   = w + (size_t)1  * MSZ;
  float* X4   = w + (size_t)2  * MSZ;
  float* X6   = w + (size_t)3  * MSZ;
  float* PR   = w + (size_t)4  * MSZ;   // cos(X)
  float* S0   = w + (size_t)5  * MSZ;
  float* PIm  = w + (size_t)6  * MSZ;   // -sin(X)
  float* E0R  = w + (size_t)7  * MSZ;
  float* E0I  = w + (size_t)8  * MSZ;
  float* E1R  = w + (size_t)9  * MSZ;
  float* E1I  = w + (size_t)10 * MSZ;
  float* RHO0 = w + (size_t)11 * MSZ;
  float* QR   = w + (size_t)12  * MSZ;              // 64 matrices
  float* QI   = w + (size_t)(12 + CHUNKS)   * MSZ;  // 64
  float* TRb  = w + (size_t)(12 + 2*CHUNKS) * MSZ;  // 64 (reused as A1 scratch)
  float* TIb  = w + (size_t)(12 + 3*CHUNKS) * MSZ;  // 64 (reused as A2 scratch)
  float* Rb   = w + (size_t)(12 + 4*CHUNKS) * MSZ;  // 64
  float* Jb   = w + (size_t)(12 + 5*CHUNKS) * MSZ;  // 64
  float* TRC  = w + (size_t)(12 + 6*CHUNKS) * MSZ;  // 2*4096 traces

  const int evSmem  = (2*MSZ + 512) * (int)sizeof(float);   // 133 KB
  const int fftSmem = 4 * 8192 * (int)sizeof(float);        // 128 KB
  (void)hipFuncSetAttribute(reinterpret_cast<const void*>(evolve_kernel),
                            hipFuncAttributeMaxDynamicSharedMemorySize, evSmem);
  (void)hipFuncSetAttribute(reinterpret_cast<const void*>(fft_kernel),
                            hipFuncAttributeMaxDynamicSharedMemorySize, fftSmem);

  // 1) X = DT*H0 (bit tricks), rho0 = IHx
  build_kernel<<<64, 256, 0, stream>>>(X, RHO0, h_mat);

  // 2) Taylor: cos/sin of X (||X|| ~ 0.02 -> 3 matmuls suffice)
  mm128_kernel<<<8, 256, 0, stream>>>(X2, X,  X,  1.f, 0.f);
  mm128_kernel<<<8, 256, 0, stream>>>(X4, X2, X2, 1.f, 0.f);
  mm128_kernel<<<8, 256, 0, stream>>>(X6, X4, X2, 1.f, 0.f);
  build_cs_kernel<<<64, 256, 0, stream>>>(PR, S0, X2, X4, X6);
  mm128_kernel<<<8, 256, 0, stream>>>(PIm, X, S0, -1.f, 0.f);   // PI = -sin(X)

  // 3) P^64 via 6 complex squarings
  cmm128_kernel<<<8, 256, 0, stream>>>(E0R, E0I, PR,  PIm, PR,  PIm);  // P^2
  cmm128_kernel<<<8, 256, 0, stream>>>(E1R, E1I, E0R, E0I, E0R, E0I);  // P^4
  cmm128_kernel<<<8, 256, 0, stream>>>(E0R, E0I, E1R, E1I, E1R, E1I);  // P^8
  cmm128_kernel<<<8, 256, 0, stream>>>(E1R, E1I, E0R, E0I, E0R, E0I);  // P^16
  cmm128_kernel<<<8, 256, 0, stream>>>(E0R, E0I, E1R, E1I, E1R, E1I);  // P^32
  cmm128_kernel<<<8, 256, 0, stream>>>(E1R, E1I, E0R, E0I, E0R, E0I);  // P^64

  // 4) chunk propagators Q_j = P^(64 j), Q_0 = I
  ident_kernel<<<64, 256, 0, stream>>>(QR, QI);
  for (int j = 1; j < CHUNKS; ++j)
    cmm128_kernel<<<8, 256, 0, stream>>>(QR + (size_t)j*MSZ, QI + (size_t)j*MSZ,
                                         QR + (size_t)(j-1)*MSZ, QI + (size_t)(j-1)*MSZ,
                                         E1R, E1I);

  // 5) seed 64 chains: rho_{64j} = Q_j rho0 Q_j^dag
  init1_kernel<<<dim3(8, CHUNKS), 256, 0, stream>>>(TRb, TIb, QR, QI, RHO0);
  init2_kernel<<<dim3(8, CHUNKS), 256, 0, stream>>>(Rb, Jb, TRb, TIb, QR, QI);

  // 6) 64 parallel chains x 64 steps; traces recorded per step
  evolve_kernel<<<CHUNKS, 256, evSmem, stream>>>(PR, PIm, Rb, Jb, TRb, TIb, TRC);

  // 7) FID, time, freq; then zero-padded 8192-pt FFT + fftshift -> FTspec
  post_kernel<<<32, 256, 0, stream>>>(TRC, out);
  fft_kernel<<<1, 256, fftSmem, stream>>>(out, out + 2*NSAMP + NSAMP);
}